// MetaConv1d_54322746359864
// MI455X (gfx1250) — compile-verified
//
#include <hip/hip_runtime.h>
#include <hip/hip_bf16.h>

typedef __attribute__((ext_vector_type(2))) float v2f;
typedef __attribute__((ext_vector_type(8))) float v8f;
typedef __attribute__((ext_vector_type(4))) unsigned u32x4;
typedef __attribute__((ext_vector_type(4))) int      i32x4;
typedef __attribute__((ext_vector_type(8))) int      i32x8;

#define BN_TOT 6624      // B*N
#define IN_CH  64
#define OUT_CH 64
#define KW     3
#define LL     128
#define LOUT   126
#define MDIM   32
#define OKDIM  192       // OUT_CH*KW

// ---- LDS layout (float offsets), row strides padded against bank conflicts
#define OFF_META 0                    // 32
#define OFF_BIAS 32                   // 64
#define OFF_HID  96                   // 64 x 33   hidden[c][m]
#define HID_S    33
#define OFF_W2   2208                 // 192 x 33  w2_w[ok][m]
#define W2_S     33
#define OFF_W2O  8544                 // 64 x 196  W2out[c][ok]
#define W2O_S    196
#define OFF_X    21088                // 130 x 65  x[l][c]; col 64 = TDM pad, rows 128..129 zeroed
#define X_S      65
#define OFF_OUT  29538                // 126 x 65  out[t][o] staging for coalesced store
#define OUT_S    65
#define SMEM_FLOATS 37728             // 150912 bytes

#if __has_builtin(__builtin_amdgcn_tensor_load_to_lds) && __has_builtin(__builtin_amdgcn_s_wait_tensorcnt)
#define USE_TDM 1
#else
#define USE_TDM 0
#endif

__global__ __launch_bounds__(256) void metaconv1d_kernel(
    const float* __restrict__ meta, const float* __restrict__ input,
    const float* __restrict__ w1_w, const float* __restrict__ w1_b,
    const float* __restrict__ w2_w, const float* __restrict__ w2_b,
    const float* __restrict__ bl_w, const float* __restrict__ bl_b,
    float* __restrict__ out)
{
    extern __shared__ float sm[];
    const int bn    = blockIdx.x;
    const int tid   = threadIdx.x;
    const int lane  = tid & 31;
    const int wave  = tid >> 5;
    const int ln    = lane & 15;          // N (or M-for-A) index within tile
    const int half2 = (lane >> 4) << 1;   // K sub-offset for A/B fragments
    const int half8 = (lane >> 4) << 3;   // M sub-offset for C/D fragments

    // ---------------- Phase A0: kick off TDM DMA of x[128][64] -> LDS stride-65 ----------------
#if USE_TDM
    if (wave == 0) {
        const uint64_t ga = (uint64_t)(const void*)(input + (size_t)bn * (LL * IN_CH));
        const unsigned ldsb = (unsigned)(size_t)(&sm[OFF_X]);   // generic ptr low 32 bits = LDS offset
        u32x4 g0;
        g0[0] = 1u;                                       // count=1, is_restore=0, gather=0
        g0[1] = ldsb;                                     // lds_addr
        g0[2] = (unsigned)(ga & 0xffffffffu);             // global_addr[31:0]
        g0[3] = (unsigned)((ga >> 32) & 0x01ffffffu)      // global_addr[56:32]
              | (2u << 30);                               // type=2 (image)
        i32x8 g1;
        g1[0] = (2 << 16)      // data_size = 4B
              | (1 << 20)      // pad_enable
              | (5 << 22)      // pad_interval: 64 DWORDs
              | (0 << 25);     // pad_amount: 1 DWORD  -> stride-65 rows in LDS
        g1[1] = 0x20000000;    // tensor_dim0 = 8192 (bits 79:48, low half)
        g1[2] = 0x00010000;    // tensor_dim0 hi = 0, tensor_dim1 = 1
        g1[3] = 0x20000000;    // tensor_dim1 hi = 0, tile_dim0 = 8192
        g1[4] = 1;             // tile_dim1 = 1, tile_dim2 = 0
        g1[5] = 8192;          // tensor_dim0_stride lo
        g1[6] = 0;
        g1[7] = 0;
        i32x4 z4 = {0, 0, 0, 0};
#if defined(__clang_major__) && __clang_major__ >= 23
        i32x8 z8 = {0, 0, 0, 0, 0, 0, 0, 0};
        __builtin_amdgcn_tensor_load_to_lds(g0, g1, z4, z4, z8, 0);
#else
        __builtin_amdgcn_tensor_load_to_lds(g0, g1, z4, z4, 0);
#endif
    }
#endif

    // ---------------- Phase A: stage meta, w2_w (and x if no TDM) ----------------
    if (tid < MDIM) sm[OFF_META + tid] = meta[(size_t)bn * MDIM + tid];
    for (int i = tid; i < OKDIM * MDIM; i += 256)
        sm[OFF_W2 + (i >> 5) * W2_S + (i & 31)] = w2_w[i];
#if !USE_TDM
    {
        const float* xin = input + (size_t)bn * (LL * IN_CH);
        for (int i = tid; i < LL * IN_CH; i += 256) {
            int l = i >> 6, c = i & 63;
            sm[OFF_X + l * X_S + c] = xin[i];          // natural [l][c], stride 65
        }
    }
#endif
    if (tid < 2 * X_S)                                  // zero rows l = 128,129 (TDM never touches)
        sm[OFF_X + LL * X_S + tid] = 0.0f;
    __syncthreads();

    // ---------------- Phase B: hidden = meta @ w1_w^T + w1_b ; bias ----------------
    for (int idx = tid; idx < IN_CH * MDIM; idx += 256) {
        if (idx + 256 < IN_CH * MDIM)
            __builtin_prefetch(w1_w + (size_t)(idx + 256) * MDIM, 0, 0);
        float acc = w1_b[idx];
        const float* row = w1_w + (size_t)idx * MDIM;
        #pragma unroll
        for (int d = 0; d < MDIM; ++d) acc += sm[OFF_META + d] * row[d];
        sm[OFF_HID + (idx >> 5) * HID_S + (idx & 31)] = acc;
    }
    if (tid < OUT_CH) {
        float acc = bl_b[tid];
        const float* row = bl_w + (size_t)tid * MDIM;
        #pragma unroll
        for (int d = 0; d < MDIM; ++d) acc += sm[OFF_META + d] * row[d];
        sm[OFF_BIAS + tid] = acc;
    }
    __syncthreads();

    // ---------------- Phase C: W2out(64x192) = hidden(64x32) @ w2_w^T + w2_b ----------------
    // WMMA f32 16x16x4 ; 4x12 tiles over 8 waves (overlaps the TDM transfer)
    for (int tile = wave; tile < 4 * 12; tile += 8) {
        const int cm = (tile / 12) * 16;   // M = c
        const int nn = (tile % 12) * 16;   // N = ok
        const float binit = w2_b[nn + ln]; // bias depends only on N -> same in all 8 regs
        v8f acc;
        #pragma unroll
        for (int v = 0; v < 8; ++v) acc[v] = binit;
        #pragma unroll
        for (int kk = 0; kk < MDIM; kk += 4) {
            v2f a, b;
            const int arow = OFF_HID + (cm + ln) * HID_S + kk + half2;
            a.x = sm[arow];
            a.y = sm[arow + 1];
            const int brow = OFF_W2 + (nn + ln) * W2_S + kk + half2;  // B[k,n]=w2_w[n][k]
            b.x = sm[brow];
            b.y = sm[brow + 1];
            acc = __builtin_amdgcn_wmma_f32_16x16x4_f32(
                false, a, false, b, (short)0, acc, false, false);
        }
        #pragma unroll
        for (int v = 0; v < 8; ++v)
            sm[OFF_W2O + (cm + v + half8) * W2O_S + nn + ln] = acc[v];
    }
#if USE_TDM
    __builtin_amdgcn_s_wait_tensorcnt(0);   // wave0's DMA done (no-op for other waves)
#endif
    __syncthreads();

    // ---------------- Phase D: conv  out(64 x 126) = A(64x192) @ B(192x128pad) ----------------
    // k = j*64 + c ;  A[o,k] = W2out[c][o*3+j] ;  B[k,t] = x[t+j][c]
    for (int tile = wave; tile < 4 * 8; tile += 8) {
        const int m0 = (tile >> 3) * 16;   // o base
        const int t0 = (tile & 7) * 16;    // t base
        v8f acc;
        #pragma unroll
        for (int v = 0; v < 8; ++v) acc[v] = sm[OFF_BIAS + m0 + v + half8];
        const int o = m0 + ln;
        #pragma unroll 4
        for (int kk = 0; kk < OKDIM; kk += 4) {
            const int j  = kk >> 6;
            const int cb = (kk & 63) + half2;
            v2f a, b;
            a.x = sm[OFF_W2O + cb       * W2O_S + o * 3 + j];
            a.y = sm[OFF_W2O + (cb + 1) * W2O_S + o * 3 + j];
            const int xrow = OFF_X + (t0 + ln + j) * X_S + cb;
            b.x = sm[xrow];
            b.y = sm[xrow + 1];
            acc = __builtin_amdgcn_wmma_f32_16x16x4_f32(
                false, a, false, b, (short)0, acc, false, false);
        }
        const int t = t0 + ln;
        if (t < LOUT) {
            #pragma unroll
            for (int v = 0; v < 8; ++v)
                sm[OFF_OUT + t * OUT_S + m0 + v + half8] = acc[v];
        }
    }
    __syncthreads();

    // ---------------- Phase E: coalesced store out[bn, t, o] ----------------
    float* obase = out + (size_t)bn * (LOUT * OUT_CH);
    for (int i = tid; i < LOUT * OUT_CH; i += 256)
        obase[i] = sm[OFF_OUT + (i >> 6) * OUT_S + (i & 63)];
}

extern "C" void kernel_launch(void* const* d_in, const int* in_sizes, int n_in,
                              void* d_out, int out_size, void* d_ws, size_t ws_size,
                              hipStream_t stream) {
    const float* meta  = (const float*)d_in[0];
    const float* input = (const float*)d_in[1];
    const float* w1_w  = (const float*)d_in[2];
    const float* w1_b  = (const float*)d_in[3];
    const float* w2_w  = (const float*)d_in[4];
    const float* w2_b  = (const float*)d_in[5];
    const float* bl_w  = (const float*)d_in[6];
    const float* bl_b  = (const float*)d_in[7];
    float* outp = (float*)d_out;

    dim3 grid(BN_TOT), block(256);
    size_t smem = SMEM_FLOATS * sizeof(float);
    hipLaunchKernelGGL(metaconv1d_kernel, grid, block, smem, stream,
                       meta, input, w1_w, w1_b, w2_w, w2_b, bl_w, bl_b, outp);
}